// GNN_68049461837969
// MI455X (gfx1250) — compile-verified
//
#include <hip/hip_runtime.h>

// Problem constants from the reference
#define B_ 8
#define N_ 20000
#define F_ 128
#define E_ 320000
static constexpr int ROWS = B_ * N_;          // 160000 rows of the (B*N, F) activation matrix
static constexpr float EPS_ = 1e-5f;

typedef float v2f __attribute__((ext_vector_type(2)));
typedef float v8f __attribute__((ext_vector_type(8)));

// ---------------------------------------------------------------------------
// Graph preprocessing (layer-invariant): degree -> dinv -> per-edge norm
// ---------------------------------------------------------------------------
__global__ __launch_bounds__(256) void gnn_deg_kernel(const int* __restrict__ tgt,
                                                      float* __restrict__ deg) {
    int e = blockIdx.x * 256 + threadIdx.x;
    if (e < E_) unsafeAtomicAdd(&deg[tgt[e]], 1.0f);
}

__global__ __launch_bounds__(256) void gnn_dinv_kernel(float* __restrict__ deg_dinv) {
    int i = blockIdx.x * 256 + threadIdx.x;
    if (i < N_) deg_dinv[i] = rsqrtf(deg_dinv[i] + 1.0f);   // +1 self-loop, deg>=1 always
}

__global__ __launch_bounds__(256) void gnn_norm_kernel(const int* __restrict__ src,
                                                       const int* __restrict__ tgt,
                                                       const float* __restrict__ dinv,
                                                       float* __restrict__ norm) {
    int e = blockIdx.x * 256 + threadIdx.x;
    if (e < E_) norm[e] = dinv[src[e]] * dinv[tgt[e]];
}

// ---------------------------------------------------------------------------
// FP32 GEMM via V_WMMA_F32_16X16X4_F32.
//   out[M x 128] = A[M x 128] @ W[128 x 128] (+ bias)
// Block = 256 threads (8 waves). W staged in LDS (64KB) with an XOR swizzle:
// column bit4 is XORed with row bit1, so the two wave-halves (reading rows
// R,R+1 vs R+2,R+3) hit disjoint 16-bank groups -> conflict-free ds loads.
// Each wave computes one 16-row x 128-col strip: 32 K-chunks x 8 col tiles.
// ---------------------------------------------------------------------------
__global__ __launch_bounds__(256) void gnn_gemm_wmma(const float* __restrict__ A,
                                                     const float* __restrict__ W,
                                                     const float* __restrict__ bias,
                                                     float* __restrict__ out) {
    __shared__ float sW[F_ * F_];                 // exactly 64 KB

    int tid = threadIdx.x;
    // Cooperative stage of W: 128 rows x 32 float4 per row, XOR-swizzled.
    for (int i = tid; i < F_ * (F_ / 4); i += 256) {
        int row = i >> 5;
        int c4  = (i & 31) << 2;                  // column of this float4
        float4 v = ((const float4*)W)[i];
        int sc = c4 ^ ((row & 2) << 3);           // flip bit4 of col by bit1 of row
        float* dst = &sW[row * F_ + sc];
        dst[0] = v.x; dst[1] = v.y; dst[2] = v.z; dst[3] = v.w;
    }
    __syncthreads();

    const int lane = tid & 31;
    const int wave = tid >> 5;
    const int half = lane >> 4;                   // 0: lanes 0-15, 1: lanes 16-31
    const int l16  = lane & 15;
    const int kb   = half * 2;                    // A/B fragment K offset per half
    const int m0   = (blockIdx.x * 8 + wave) * 16;

    const float* arow = A + (size_t)(m0 + l16) * F_ + kb;

    v8f acc[8] = {};                              // 8 column tiles x 16x16 f32 accum

    for (int kc = 0; kc < 32; ++kc) {             // K = 128 in chunks of 4
        v2f a = *(const v2f*)(arow + kc * 4);     // A frag: [m][k..k+1] per lane
        const int r0 = kc * 4 + kb;
#pragma unroll
        for (int t = 0; t < 8; ++t) {
            int n = t * 16 + l16;
            v2f b;
            b.x = sW[(r0    ) * F_ + (n ^ (((r0    ) & 2) << 3))];
            b.y = sW[(r0 + 1) * F_ + (n ^ (((r0 + 1) & 2) << 3))];
            acc[t] = __builtin_amdgcn_wmma_f32_16x16x4_f32(
                false, a, false, b, (short)0, acc[t], false, false);
        }
    }

    // C/D layout: VGPR r -> rows (half*8 + r), col = l16 within tile
    const int rbase = half * 8;
    for (int t = 0; t < 8; ++t) {
        int ncol = t * 16 + l16;
        float bv = bias ? bias[ncol] : 0.0f;
        float* op = out + (size_t)(m0 + rbase) * F_ + ncol;
#pragma unroll
        for (int r = 0; r < 8; ++r)
            op[(size_t)r * F_] = acc[t][r] + bv;
    }
}

// ---------------------------------------------------------------------------
// Edge scatter: agg[:, tgt] += h[:, src] * norm  (self-loops inline at e>=E)
// One block per edge, 128 lanes = feature dim -> coalesced loads + atomics.
// ---------------------------------------------------------------------------
__global__ __launch_bounds__(128) void gnn_scatter(const float* __restrict__ h,
                                                   const int* __restrict__ src,
                                                   const int* __restrict__ tgt,
                                                   const float* __restrict__ norm,
                                                   const float* __restrict__ dinv,
                                                   float* __restrict__ agg) {
    int e = blockIdx.x;
    int f = threadIdx.x;
    int s, t; float w;
    if (e < E_) { s = src[e]; t = tgt[e]; w = norm[e]; }
    else        { s = t = e - E_; float d = dinv[s]; w = d * d; }
#pragma unroll
    for (int b = 0; b < B_; ++b) {
        float v = h[((size_t)b * N_ + s) * F_ + f] * w;
        unsafeAtomicAdd(&agg[((size_t)b * N_ + t) * F_ + f], v);
    }
}

// ---------------------------------------------------------------------------
// BatchNorm statistics: per-feature sum / sumsq over B*N rows.
// stats[0:128]=sum, [128:256]=sumsq, [256:384]=scale, [384:512]=shift
// ---------------------------------------------------------------------------
__global__ __launch_bounds__(128) void gnn_bn_stats(const float* __restrict__ agg,
                                                    float* __restrict__ stats) {
    int f = threadIdx.x;
    float s = 0.0f, s2 = 0.0f;
    for (int r = blockIdx.x; r < ROWS; r += gridDim.x) {
        float v = agg[(size_t)r * F_ + f];
        s += v; s2 += v * v;
    }
    unsafeAtomicAdd(&stats[f], s);
    unsafeAtomicAdd(&stats[F_ + f], s2);
}

__global__ __launch_bounds__(128) void gnn_bn_finalize(const float* __restrict__ gamma,
                                                       const float* __restrict__ beta,
                                                       float* __restrict__ stats) {
    int f = threadIdx.x;
    const float inv = 1.0f / (float)ROWS;
    float mean = stats[f] * inv;
    float var  = stats[F_ + f] * inv - mean * mean;
    float sc   = gamma[f] * rsqrtf(var + EPS_);
    stats[2 * F_ + f] = sc;
    stats[3 * F_ + f] = beta[f] - mean * sc;
}

// out = relu(agg*scale + shift) + prev, vectorized float4
__global__ __launch_bounds__(256) void gnn_bn_apply(const float* __restrict__ agg,
                                                    const float* __restrict__ prev,
                                                    const float* __restrict__ stats,
                                                    float* __restrict__ out) {
    const float4* a4  = (const float4*)agg;
    const float4* p4  = (const float4*)prev;
    const float4* sc4 = (const float4*)(stats + 2 * F_);
    const float4* sh4 = (const float4*)(stats + 3 * F_);
    float4* o4 = (float4*)out;
    const int total = ROWS * (F_ / 4);            // 5,120,000
    for (int i = blockIdx.x * 256 + threadIdx.x; i < total; i += gridDim.x * 256) {
        int fq = i & 31;                          // float4 index within feature dim
        float4 a = a4[i], p = p4[i], sc = sc4[fq], sh = sh4[fq];
        float4 r;
        r.x = fmaxf(fmaf(a.x, sc.x, sh.x), 0.0f) + p.x;
        r.y = fmaxf(fmaf(a.y, sc.y, sh.y), 0.0f) + p.y;
        r.z = fmaxf(fmaf(a.z, sc.z, sh.z), 0.0f) + p.z;
        r.w = fmaxf(fmaf(a.w, sc.w, sh.w), 0.0f) + p.w;
        o4[i] = r;
    }
}

// ---------------------------------------------------------------------------
extern "C" void kernel_launch(void* const* d_in, const int* in_sizes, int n_in,
                              void* d_out, int out_size, void* d_ws, size_t ws_size,
                              hipStream_t stream) {
    (void)in_sizes; (void)n_in; (void)out_size; (void)ws_size;

    const float* x      = (const float*)d_in[0];
    const int*   eidx   = (const int*)  d_in[1];   // [2][E]
    const float* Ws     = (const float*)d_in[2];   // [3][128][128]
    const float* gammas = (const float*)d_in[3];   // [3][128]
    const float* betas  = (const float*)d_in[4];   // [3][128]
    const float* W_out  = (const float*)d_in[5];   // [128][128]
    const float* b_out  = (const float*)d_in[6];   // [128]
    float* out = (float*)d_out;

    const int* src = eidx;
    const int* tgt = eidx + E_;

    const size_t BNF = (size_t)ROWS * F_;          // 20.48M floats
    float* P0    = (float*)d_ws;                   // activation ping
    float* P1    = P0 + BNF;                       // activation pong
    float* AGG   = P1 + BNF;                       // scatter accumulator
    float* dinv  = AGG + BNF;                      // N floats (deg, then rsqrt in place)
    float* norm  = dinv + N_;                      // E floats
    float* stats = norm + E_;                      // 512 floats

    // Graph preprocessing (once per launch; layer-invariant)
    hipMemsetAsync(dinv, 0, N_ * sizeof(float), stream);
    gnn_deg_kernel <<<(E_ + 255) / 256, 256, 0, stream>>>(tgt, dinv);
    gnn_dinv_kernel<<<(N_ + 255) / 256, 256, 0, stream>>>(dinv);
    gnn_norm_kernel<<<(E_ + 255) / 256, 256, 0, stream>>>(src, tgt, dinv, norm);

    const float* prev = x;
    float* hbuf  = P0;
    float* other = P1;
    for (int l = 0; l < 3; ++l) {
        // h = prev @ Ws[l]
        gnn_gemm_wmma<<<ROWS / 128, 256, 0, stream>>>(prev, Ws + (size_t)l * F_ * F_,
                                                      nullptr, hbuf);
        // agg = scatter_add(h[:,src] * norm)
        hipMemsetAsync(AGG, 0, BNF * sizeof(float), stream);
        gnn_scatter<<<E_ + N_, 128, 0, stream>>>(hbuf, src, tgt, norm, dinv, AGG);
        // batchnorm stats + fused relu/residual apply (overwrites dead h buffer)
        hipMemsetAsync(stats, 0, 2 * F_ * sizeof(float), stream);
        gnn_bn_stats   <<<1024, 128, 0, stream>>>(AGG, stats);
        gnn_bn_finalize<<<1,    128, 0, stream>>>(gammas + l * F_, betas + l * F_, stats);
        gnn_bn_apply   <<<2048, 256, 0, stream>>>(AGG, prev, stats, hbuf);

        prev = hbuf;
        float* tmp = hbuf; hbuf = other; other = tmp;   // ping-pong
    }

    // out = prev @ W_out + b_out
    gnn_gemm_wmma<<<ROWS / 128, 256, 0, stream>>>(prev, W_out, b_out, out);
}